// LogEig_25778393711092
// MI455X (gfx1250) — compile-verified
//
#include <hip/hip_runtime.h>
#include <math.h>

// ---------------------------------------------------------------------------
// Batched SPD matrix logarithm, 32768 x (32x32) fp32, for MI455X (gfx1250).
// logm(A) = log(||A||_F)*I + log(A/||A||_F), computed with an inverse
// scaling-and-squaring scheme built entirely from 32x32x32 fp32 GEMMs that
// run on V_WMMA_F32_16X16X4_F32. One matrix per wave32, LDS-resident.
// ---------------------------------------------------------------------------

typedef __attribute__((ext_vector_type(2))) float v2f;
typedef __attribute__((ext_vector_type(4))) float v4f;
typedef __attribute__((ext_vector_type(8))) float v8f;

#define LDSTRIDE 34          // padded row stride: bank-conflict-free columns,
                             // 8B-aligned (k,k+1) pairs for A-fragments
#define MAT_ELEMS 1024       // 32*32
#define WAVES_PER_BLOCK 4

// ---- WMMA fragment loaders (layouts per CDNA5 ISA 7.12.2) ------------------
// A-matrix 16x4 fp32: lane m = M row, K = 2*lanegroup + vgpr  (contiguous pair)
__device__ __forceinline__ v2f load_a_frag(const float* M, int ti, int kc,
                                           int g, int m) {
    const int row = 16 * ti + m;
    const int k  = 4 * kc + 2 * g;
    v2f a;
    a.x = M[row * LDSTRIDE + k];
    a.y = M[row * LDSTRIDE + k + 1];
    return a;
}

// B-matrix 4x16 fp32: lane m = N col, K = 2*lanegroup + vgpr
__device__ __forceinline__ v2f load_b_frag(const float* M, int tj, int kc,
                                           int g, int m) {
    const int col = 16 * tj + m;
    const int k  = 4 * kc + 2 * g;
    v2f b;
    b.x = M[(k)     * LDSTRIDE + col];
    b.y = M[(k + 1) * LDSTRIDE + col];
    return b;
}

// C/D 16x16 fp32: vgpr r, lanegroup g -> row = r + 8g, col = lane&15
__device__ __forceinline__ void store_tile(float* D, v8f acc, int ti, int tj,
                                           float alpha, float diag,
                                           int g, int m) {
#pragma unroll
    for (int r = 0; r < 8; ++r) {
        const int row = 16 * ti + r + 8 * g;
        const int col = 16 * tj + m;
        float v = alpha * acc[r];
        if (row == col) v += diag;
        D[row * LDSTRIDE + col] = v;
    }
}

// D = alpha*(A x B) + diag*I, all 32x32 in LDS (stride LDSTRIDE).
// 2x2 tiles of 16x16, K = 32 as 8 chunks of 4  ->  32 wmma per GEMM.
__device__ __forceinline__ void mm_step(const float* A, const float* Bm,
                                        float* D, float alpha, float diag,
                                        int g, int m) {
    v8f acc00 = {0.f, 0.f, 0.f, 0.f, 0.f, 0.f, 0.f, 0.f};
    v8f acc01 = acc00, acc10 = acc00, acc11 = acc00;
#pragma unroll
    for (int kc = 0; kc < 8; ++kc) {
        const v2f a0 = load_a_frag(A, 0, kc, g, m);
        const v2f a1 = load_a_frag(A, 1, kc, g, m);
        const v2f b0 = load_b_frag(Bm, 0, kc, g, m);
        const v2f b1 = load_b_frag(Bm, 1, kc, g, m);
        acc00 = __builtin_amdgcn_wmma_f32_16x16x4_f32(false, a0, false, b0,
                                                      (short)0, acc00, false, false);
        acc01 = __builtin_amdgcn_wmma_f32_16x16x4_f32(false, a0, false, b1,
                                                      (short)0, acc01, false, false);
        acc10 = __builtin_amdgcn_wmma_f32_16x16x4_f32(false, a1, false, b0,
                                                      (short)0, acc10, false, false);
        acc11 = __builtin_amdgcn_wmma_f32_16x16x4_f32(false, a1, false, b1,
                                                      (short)0, acc11, false, false);
    }
    store_tile(D, acc00, 0, 0, alpha, diag, g, m);
    store_tile(D, acc01, 0, 1, alpha, diag, g, m);
    store_tile(D, acc10, 1, 0, alpha, diag, g, m);
    store_tile(D, acc11, 1, 1, alpha, diag, g, m);
}

__global__ __launch_bounds__(32 * WAVES_PER_BLOCK) void
logm_spd_kernel(const float* __restrict__ in, float* __restrict__ out,
                int nmat) {
    __shared__ float lds[WAVES_PER_BLOCK][3][32 * LDSTRIDE];

    const int wave = threadIdx.x >> 5;
    const int lane = threadIdx.x & 31;
    const int g = lane >> 4;      // lanegroup (0: lanes 0-15, 1: lanes 16-31)
    const int m = lane & 15;

    const int mat = blockIdx.x * WAVES_PER_BLOCK + wave;
    if (mat >= nmat) return;      // wave-uniform: EXEC stays all-ones for WMMA

    float* b0 = &lds[wave][0][0]; // Y
    float* b1 = &lds[wave][1][0]; // Z / X
    float* b2 = &lds[wave][2][0]; // T / P

    // ---- Stream matrix in (coalesced float4) + Frobenius norm --------------
    const float* src = in + (size_t)mat * MAT_ELEMS;
    v4f stash[8];
    float ss = 0.f;
#pragma unroll
    for (int c = 0; c < 8; ++c) {
        const int f = c * 32 + lane;                 // float4 index, contiguous
        const v4f x = ((const v4f*)src)[f];
        stash[c] = x;
        ss += x.x * x.x + x.y * x.y + x.z * x.z + x.w * x.w;
    }
#pragma unroll
    for (int off = 16; off > 0; off >>= 1)
        ss += __shfl_xor(ss, off, 32);

    const float scale = rsqrtf(ss);                  // 1/||A||_F
    const float logs  = 0.5f * logf(ss);             // log(||A||_F)

    // ---- b0 = B = A / ||A||_F  (eigenvalues in (0,1]) ----------------------
#pragma unroll
    for (int c = 0; c < 8; ++c) {
        const int f = c * 32 + lane;
        const int row = f >> 3;
        const int col = (f & 7) * 4;
        float* d = &b0[row * LDSTRIDE + col];
        d[0] = stash[c].x * scale;
        d[1] = stash[c].y * scale;
        d[2] = stash[c].z * scale;
        d[3] = stash[c].w * scale;
    }

    // ---- Inverse scaling-and-squaring: Y <- B^(1/16) -----------------------
    // Coupled inverse-free Newton-Schulz sqrt:
    //   T = 1.5 I - 0.5 Z*Y ;  Y <- Y*T ;  Z <- T*Z
    // Converges since spec(B) in (0,1]. Iteration counts sized for
    // r0 = ||I - B|| up to ~0.95 (first round), shrinking each round.
#pragma unroll 1
    for (int sq = 0; sq < 4; ++sq) {
#pragma unroll
        for (int c = 0; c < 32; ++c)                 // Z = I (lane owns a row)
            b1[lane * LDSTRIDE + c] = (lane == c) ? 1.f : 0.f;

        const int iters = (sq == 0) ? 9 : (sq == 1) ? 7 : (sq == 2) ? 6 : 5;
#pragma unroll 1
        for (int it = 0; it < iters; ++it) {
            mm_step(b1, b0, b2, -0.5f, 1.5f, g, m);  // T = 1.5I - 0.5 Z*Y
            mm_step(b0, b2, b0,  1.0f, 0.0f, g, m);  // Y = Y*T
            mm_step(b2, b1, b1,  1.0f, 0.0f, g, m);  // Z = T*Z
        }
    }

    // ---- log(B) = 16 * log(Y),  log(Y) via Mercator series, X = Y - I ------
    // ||X|| <~ 0.2  ->  degree-12 series is fp32-exact.
#pragma unroll
    for (int c = 0; c < 32; ++c) {
        const float y = b0[lane * LDSTRIDE + c];
        const float x = y - ((lane == c) ? 1.f : 0.f);
        b1[lane * LDSTRIDE + c] = x;                        // X
        b2[lane * LDSTRIDE + c] = (-1.f / 12.f) * x +       // P = c12*X + c11*I
                                  ((lane == c) ? (1.f / 11.f) : 0.f);
    }
#pragma unroll 1
    for (int k = 10; k >= 2; --k) {
        const float ck = ((k & 1) ? 1.f : -1.f) / (float)k;
        mm_step(b1, b2, b2, 1.f, ck, g, m);          // P = X*P + ck*I
    }
    mm_step(b1, b2, b2, 1.f, 1.f, g, m);             // P = X*P + c1*I  (c1=1)
    // log(A) = 16 * X*P + log(s)*I   (fold 2^4 from the 4 sqrt rounds)
    mm_step(b1, b2, b2, 16.f, logs, g, m);

    // ---- Stream result out (coalesced float4) ------------------------------
    float* dst = out + (size_t)mat * MAT_ELEMS;
#pragma unroll
    for (int c = 0; c < 8; ++c) {
        const int f = c * 32 + lane;
        const int row = f >> 3;
        const int col = (f & 7) * 4;
        const float* s = &b2[row * LDSTRIDE + col];
        v4f x;
        x.x = s[0]; x.y = s[1]; x.z = s[2]; x.w = s[3];
        ((v4f*)dst)[f] = x;
    }
}

extern "C" void kernel_launch(void* const* d_in, const int* in_sizes, int n_in,
                              void* d_out, int out_size, void* d_ws,
                              size_t ws_size, hipStream_t stream) {
    (void)n_in; (void)d_ws; (void)ws_size; (void)out_size;
    const float* a = (const float*)d_in[0];
    float* outp = (float*)d_out;
    const int nmat = in_sizes[0] / MAT_ELEMS;        // 32768
    const int blocks = (nmat + WAVES_PER_BLOCK - 1) / WAVES_PER_BLOCK;
    logm_spd_kernel<<<blocks, 32 * WAVES_PER_BLOCK, 0, stream>>>(a, outp, nmat);
}